// HybridVQC_TorchModule_70712341561595
// MI455X (gfx1250) — compile-verified
//
#include <hip/hip_runtime.h>

#define NQ  12
#define NL  6
#define DIM 4096            // 2^12 amplitudes
#define TPB 256             // 8 wave32 per workgroup
#define PER (DIM / TPB)     // 16 amplitudes per thread

typedef float v2f __attribute__((ext_vector_type(2)));
typedef float v8f __attribute__((ext_vector_type(8)));

struct cplx { float re, im; };
__device__ __forceinline__ cplx cmul(cplx a, cplx b) {
  return { a.re * b.re - a.im * b.im, a.re * b.im + a.im * b.re };
}
__device__ __forceinline__ cplx cadd(cplx a, cplx b) { return { a.re + b.re, a.im + b.im }; }
__device__ __forceinline__ cplx crs(float s, cplx a) { return { s * a.re, s * a.im }; }

// U = RX(t2) * RZ(t1) * RY(t0)   (PennyLane conventions)
__device__ __forceinline__ void fused_u(float t0, float t1, float t2, cplx U[2][2]) {
  float cy, sy, cz, sz, cx, sx;
  sincosf(0.5f * t0, &sy, &cy);
  sincosf(0.5f * t1, &sz, &cz);
  sincosf(0.5f * t2, &sx, &cx);
  cplx ez0 = { cz, -sz }, ez1 = { cz, sz };   // RZ diag phases
  cplx m00 = crs( cy, ez0), m01 = crs(-sy, ez0);   // RZ*RY
  cplx m10 = crs( sy, ez1), m11 = crs( cy, ez1);
  cplx mx  = { 0.f, -sx };                         // -i*sin for RX off-diag
  U[0][0] = cadd(crs(cx, m00), cmul(mx, m10));
  U[0][1] = cadd(crs(cx, m01), cmul(mx, m11));
  U[1][0] = cadd(cmul(mx, m00), crs(cx, m10));
  U[1][1] = cadd(cmul(mx, m01), crs(cx, m11));
}

__global__ __launch_bounds__(TPB) void vqc_kernel(
    const float* __restrict__ x,    // (B, 12)
    const float* __restrict__ W,    // (6, 12, 3)
    const float* __restrict__ W1,   // (8, 1)
    const float* __restrict__ b1,   // (8,)
    const float* __restrict__ W2,   // (1, 8)
    const float* __restrict__ b2,   // (1,)
    float* __restrict__ out)        // (B,)
{
  __shared__ float sRe[DIM];
  __shared__ float sIm[DIM];
  __shared__ float sRed[TPB];

  const int tid  = threadIdx.x;
  const int b    = blockIdx.x;
  const int lane = tid & 31;
  const int wave = tid >> 5;
  const int nHi  = lane >> 4;   // half-wave: selects K-slice of A/B operands
  const int nLo  = lane & 15;

  __builtin_prefetch(&x[b * NQ], 0, 0);
  __builtin_prefetch(W, 0, 0);

  // ---- data encoding: product state  a_w = RZ(0.5x)RY(x)|0> ----
  cplx aw0[NQ], aw1[NQ];
#pragma unroll
  for (int w = 0; w < NQ; ++w) {
    float xv = x[b * NQ + w];
    float ch, sh, cq, sq;
    sincosf(0.50f * xv, &sh, &ch);
    sincosf(0.25f * xv, &sq, &cq);
    aw0[w] = { ch * cq, -ch * sq };   // cos(x/2) * e^{-ix/4}
    aw1[w] = { sh * cq,  sh * sq };   // sin(x/2) * e^{+ix/4}
  }
  for (int i = 0; i < PER; ++i) {
    int k = tid + TPB * i;
    cplx amp = { 1.f, 0.f };
#pragma unroll
    for (int w = 0; w < NQ; ++w) {
      int bit = (k >> (11 - w)) & 1;
      amp = cmul(amp, bit ? aw1[w] : aw0[w]);
    }
    sRe[k] = amp.re;
    sIm[k] = amp.im;
  }
  __syncthreads();

  // ---- variational layers ----
#pragma unroll 1
  for (int l = 0; l < NL; ++l) {
#pragma unroll 1
    for (int t = 0; t < 4; ++t) {
      // Fuse wires (3t, 3t+1, 3t+2): 8x8 complex gate -> 16x16 real matrix T,
      // applied as D = A x Tt via chains of 4 x V_WMMA_F32_16X16X4_F32.
      cplx Ua[2][2], Ub[2][2], Uc[2][2];
      const float* wl = W + (l * NQ + 3 * t) * 3;
      fused_u(wl[0], wl[1], wl[2], Ua);
      fused_u(wl[3], wl[4], wl[5], Ub);
      fused_u(wl[6], wl[7], wl[8], Uc);
      const int shift = 9 - 3 * t;              // gate bits at [shift+2 : shift]
      const int loMask = (1 << shift) - 1;

      // B operand: B[K][N] = T[N][K]; lane holds column N=nLo, K = 4c + r + 2*nHi
      v2f Bv[4];
#pragma unroll
      for (int c = 0; c < 4; ++c) {
#pragma unroll
        for (int r = 0; r < 2; ++r) {
          int q = 4 * c + r + 2 * nHi;          // K index = column of T
          int p = nLo;                          // row of T
          int pm = p & 7, qm = q & 7;
          cplx m = cmul(cmul(Ua[(pm >> 2) & 1][(qm >> 2) & 1],
                             Ub[(pm >> 1) & 1][(qm >> 1) & 1]),
                        Uc[pm & 1][qm & 1]);
          float v;
          if (p < 8) v = (q < 8) ?  m.re : -m.im;
          else       v = (q < 8) ?  m.im :  m.re;
          Bv[c][r] = v;
        }
      }

#pragma unroll 1
      for (int ch = 0; ch < 4; ++ch) {
        const int chainBase = (wave * 4 + ch) * 16;   // 16 octuple-groups / chain
        // A operand: row M=nLo = state 16-vector [Re a0..7 | Im a0..7] of group gA
        const int gA    = chainBase + nLo;
        const int baseA = ((gA >> shift) << (shift + 3)) | (gA & loMask);
        v2f Av[4];
#pragma unroll
        for (int c = 0; c < 4; ++c) {
#pragma unroll
          for (int r = 0; r < 2; ++r) {
            int K = 4 * c + r + 2 * nHi;        // K<8 iff c<2 (uniform per chunk)
            int m = K & 7;
            int idx = baseA | (m << shift);
            Av[c][r] = (c < 2) ? sRe[idx] : sIm[idx];
          }
        }
        v8f acc = {};
        acc = __builtin_amdgcn_wmma_f32_16x16x4_f32(false, Av[0], false, Bv[0], (short)0, acc, false, false);
        acc = __builtin_amdgcn_wmma_f32_16x16x4_f32(false, Av[1], false, Bv[1], (short)0, acc, false, false);
        acc = __builtin_amdgcn_wmma_f32_16x16x4_f32(false, Av[2], false, Bv[2], (short)0, acc, false, false);
        acc = __builtin_amdgcn_wmma_f32_16x16x4_f32(false, Av[3], false, Bv[3], (short)0, acc, false, false);
        // D[M][N]: M=group (r + 8*nHi within chain), N=component (nLo)
        float* dstArr = (nLo < 8) ? sRe : sIm;  // pointer select: no EXEC change
        const int mcomp = nLo & 7;
#pragma unroll
        for (int r = 0; r < 8; ++r) {
          int gg = chainBase + r + 8 * nHi;
          int baseg = ((gg >> shift) << (shift + 3)) | (gg & loMask);
          dstArr[baseg | (mcomp << shift)] = acc[r];
        }
      }
      __syncthreads();
    }

    // ---- CNOT ladder (0,1)..(10,11),(11,0) fused into one index permutation ----
    float tr[PER], ti[PER];
#pragma unroll
    for (int i = 0; i < PER; ++i) {
      int j = tid + TPB * i;
      int s = j;
      s ^= ((s >> 0) & 1) << 11;                      // CNOT(11,0)
#pragma unroll
      for (int w = 10; w >= 0; --w)                   // CNOT(w, w+1), reversed
        s ^= ((s >> (11 - w)) & 1) << (10 - w);
      tr[i] = sRe[s];
      ti[i] = sIm[s];
    }
    __syncthreads();
#pragma unroll
    for (int i = 0; i < PER; ++i) {
      int j = tid + TPB * i;
      sRe[j] = tr[i];
      sIm[j] = ti[i];
    }
    __syncthreads();
  }

  // ---- <Z> on wire 0 (index bit 11), reduce, then MLP head ----
  float part = 0.f;
  for (int i = 0; i < PER; ++i) {
    int k = tid + TPB * i;
    float p = sRe[k] * sRe[k] + sIm[k] * sIm[k];
    part += (k & 2048) ? -p : p;
  }
  sRed[tid] = part;
  __syncthreads();
  for (int s = TPB / 2; s > 0; s >>= 1) {
    if (tid < s) sRed[tid] += sRed[tid + s];
    __syncthreads();
  }
  if (tid == 0) {
    float q = sRed[0];
    float acc = b2[0];
#pragma unroll
    for (int j = 0; j < 8; ++j) {
      float h = q * W1[j] + b1[j];
      h = h > 0.f ? h : 0.f;
      acc += W2[j] * h;
    }
    out[b] = 1.f / (1.f + expf(-acc));
  }
}

extern "C" void kernel_launch(void* const* d_in, const int* in_sizes, int n_in,
                              void* d_out, int out_size, void* d_ws, size_t ws_size,
                              hipStream_t stream) {
  const float* x  = (const float*)d_in[0];
  const float* W  = (const float*)d_in[1];
  const float* W1 = (const float*)d_in[2];
  const float* b1 = (const float*)d_in[3];
  const float* W2 = (const float*)d_in[4];
  const float* b2 = (const float*)d_in[5];
  float* out = (float*)d_out;
  const int B = in_sizes[0] / NQ;   // 2048
  vqc_kernel<<<B, TPB, 0, stream>>>(x, W, W1, b1, W2, b2, out);
}